// MyViT_20753281974986
// MI455X (gfx1250) — compile-verified
//
#include <hip/hip_runtime.h>
#include <math.h>
#include <stdint.h>

// ---------------------------------------------------------------------------
// MyViT autoencoder forward for MI455X (gfx1250, wave32, WMMA f32 16x16x4,
// async global->LDS staging for attention).
// ---------------------------------------------------------------------------

typedef __attribute__((ext_vector_type(2))) float v2f;
typedef __attribute__((ext_vector_type(8))) float v8f;

// f32 WMMA: D(16x16) = A(16x4) * B(4x16) + C.
// A: lanes 0-15 row=lane, VGPRs = K0,K1; lanes 16-31 = K2,K3.
// B: lanes 0-15 col=lane, VGPRs = K0,K1; lanes 16-31 = K2,K3.
// C/D: VGPR r -> row r (lanes 0-15) / row r+8 (lanes 16-31), col = lane&15.
__device__ inline v8f wmma4(v8f acc, v2f a, v2f b) {
  return __builtin_amdgcn_wmma_f32_16x16x4_f32(
      false, a, false, b, (short)0, acc, false, false);
}

__device__ inline float gelu_f(float x) {
  return 0.5f * x * (1.0f + erff(x * 0.70710678118654752f));
}

// ---------------- positional embedding table (197 x 64) --------------------
__global__ void posemb_k(float* __restrict__ pos) {
  int idx = blockIdx.x * blockDim.x + threadIdx.x;
  if (idx >= 197 * 64) return;
  int i = idx >> 6, j = idx & 63;
  float fi = (float)i;
  if ((j & 1) == 0)
    pos[idx] = sinf(fi / powf(10000.0f, (float)j * (1.0f / 64.0f)));
  else
    pos[idx] = cosf(fi / powf(10000.0f, (float)(j - 1) * (1.0f / 64.0f)));
}

// ---------------- class-token row (row 0 of each sequence) -----------------
__global__ void cls_k(float* __restrict__ x, const float* __restrict__ ct,
                      const float* __restrict__ pos) {
  int idx = blockIdx.x * blockDim.x + threadIdx.x;
  if (idx >= 128 * 64) return;
  int n = idx >> 6, c = idx & 63;
  x[(size_t)n * 197 * 64 + c] = ct[c] + pos[c];
}

// ---------------- patch-embed GEMM: patchify fused into A loads ------------
// tokens[n, 1+p, :] = patch_row(p) @ lm_W + lm_b + posemb[1+p]
// grid (13, 128), block 128 (4 waves = 4 N-tiles of 16 over HIDDEN=64).
// Ragged rows handled by clamping the patch id; stores are masked.
__global__ void patch_embed_k(const float* __restrict__ img,
                              const float* __restrict__ W,
                              const float* __restrict__ bias,
                              const float* __restrict__ pos,
                              float* __restrict__ out) {
  int wave = threadIdx.x >> 5, lane = threadIdx.x & 31;
  int lr = lane & 15, kh = lane >> 4;
  int n = blockIdx.y;
  int row0 = blockIdx.x * 16;
  int bcol = wave * 16 + lr;
  int p = row0 + lr; if (p > 195) p = 195;  // clamp; masked at store
  const float* ab = img + (size_t)n * 200704 +
                    (size_t)((p / 14) * 32) * 448 + (p % 14) * 32;
  v8f acc = {};
  for (int r = 0; r < 32; ++r) {             // image row within patch
    const v2f* arp = (const v2f*)(ab + r * 448 + 2 * kh);
    const float* wr = W + (size_t)(r * 32 + 2 * kh) * 64 + bcol;
#pragma unroll
    for (int k8 = 0; k8 < 8; ++k8) {         // 8 wmma steps per image row
      v2f b; b.x = wr[0]; b.y = wr[64];
      acc = wmma4(acc, arp[0], b);
      arp += 2; wr += 256;
    }
  }
  float bv = bias[bcol];
#pragma unroll
  for (int r = 0; r < 8; ++r) {
    int prow = row0 + r + kh * 8;
    if (prow < 196) {
      int srow = prow + 1;
      out[((size_t)n * 197 + srow) * 64 + bcol] = acc[r] + bv + pos[srow * 64 + bcol];
    }
  }
}

// ---------------- LayerNorm over last dim (64), one wave per token ---------
__global__ void ln_k(const float* __restrict__ X, const float* __restrict__ g,
                     const float* __restrict__ b, float* __restrict__ Y, int M) {
  int wave = threadIdx.x >> 5, lane = threadIdx.x & 31;
  int t = blockIdx.x * 8 + wave;
  if (t >= M) return;
  const float* xp = X + (size_t)t * 64;
  float x0 = xp[lane], x1 = xp[lane + 32];
  float s = x0 + x1;
  for (int m = 16; m >= 1; m >>= 1) s += __shfl_xor(s, m, 32);
  float mean = s * (1.0f / 64.0f);
  float d0 = x0 - mean, d1 = x1 - mean;
  float q = d0 * d0 + d1 * d1;
  for (int m = 16; m >= 1; m >>= 1) q += __shfl_xor(q, m, 32);
  float inv = rsqrtf(q * (1.0f / 64.0f) + 1e-5f);
  Y[(size_t)t * 64 + lane]      = d0 * inv * g[lane] + b[lane];
  Y[(size_t)t * 64 + lane + 32] = d1 * inv * g[lane + 32] + b[lane + 32];
}

// ---------------- fused MSA: one workgroup per (image, head) ---------------
// Out[n,s,h*16+e] = Xres[n,s,h*16+e] + softmax(q k^T / 4) v
// x-slice staged via gfx1250 async global->LDS; Q/K/V + scores LDS-resident;
// every matmul is WMMA.
__global__ void msa_k(const float* __restrict__ Xln, const float* __restrict__ Xres,
                      const float* __restrict__ Wq, const float* __restrict__ Wk,
                      const float* __restrict__ Wv, const float* __restrict__ bq,
                      const float* __restrict__ bk, const float* __restrict__ bv,
                      float* __restrict__ Out, int S) {
  __shared__ float xh[208 * 16];   // x-slice, overwritten in-place with Q
  __shared__ float kk[208 * 16];
  __shared__ float vv[208 * 16];
  __shared__ float sc[4][16 * 208]; // per-wave score rows
  int n = blockIdx.x, h = blockIdx.y;
  int tid = threadIdx.x, wave = tid >> 5, lane = tid & 31;
  int lr = lane & 15, kh = lane >> 4;
  size_t base = (size_t)n * S * 64 + h * 16;

  // stage x-slice: async 16B global->LDS copies; zero-fill pad rows
  for (int i = tid; i < 208 * 4; i += 128) {
    int s = i >> 2, q = i & 3;
    float* dst = &xh[s * 16 + q * 4];
    if (s < S) {
      unsigned lds = (unsigned)(uintptr_t)dst;
      unsigned long long ga =
          (unsigned long long)(uintptr_t)(Xln + base + (size_t)s * 64 + q * 4);
      asm volatile("global_load_async_to_lds_b128 %0, %1, off"
                   :: "v"(lds), "v"(ga) : "memory");
    } else {
      dst[0] = 0.0f; dst[1] = 0.0f; dst[2] = 0.0f; dst[3] = 0.0f;
    }
  }
  asm volatile("s_wait_asynccnt 0x0" ::: "memory");
  __syncthreads();

  const float* wqh = Wq + h * 256;
  const float* wkh = Wk + h * 256;
  const float* wvh = Wv + h * 256;
  float bqv = bq[h * 16 + lr], bkv = bk[h * 16 + lr], bvv = bv[h * 16 + lr];

  // hoisted B-operands for the K=16 projection GEMMs
  v2f bq2[4], bk2[4], bv2[4];
#pragma unroll
  for (int k8 = 0; k8 < 4; ++k8) {
    int ka = k8 * 4 + 2 * kh;
    bq2[k8].x = wqh[ka * 16 + lr]; bq2[k8].y = wqh[(ka + 1) * 16 + lr];
    bk2[k8].x = wkh[ka * 16 + lr]; bk2[k8].y = wkh[(ka + 1) * 16 + lr];
    bv2[k8].x = wvh[ka * 16 + lr]; bv2[k8].y = wvh[(ka + 1) * 16 + lr];
  }

  // phase 1: Q,K,V tiles (row-local, K=16). Q overwrites xh rows.
  for (int t = wave; t < 13; t += 4) {
    v8f aq = {}, ak = {}, av = {};
    int ar = (t * 16 + lr) * 16 + 2 * kh;
#pragma unroll
    for (int k8 = 0; k8 < 4; ++k8) {
      v2f a = *(const v2f*)&xh[ar + k8 * 4];
      aq = wmma4(aq, a, bq2[k8]);
      ak = wmma4(ak, a, bk2[k8]);
      av = wmma4(av, a, bv2[k8]);
    }
#pragma unroll
    for (int r = 0; r < 8; ++r) {
      int rr = (t * 16 + r + kh * 8) * 16 + lr;
      xh[rr] = aq[r] + bqv;
      kk[rr] = ak[r] + bkv;
      vv[rr] = av[r] + bvv;
    }
  }
  __syncthreads();

  // phase 2: per 16-row tile: scores = q k^T / 4, softmax, o = att v
  float* scw = sc[wave];
  for (int t = wave; t < 13; t += 4) {
    int ar = (t * 16 + lr) * 16 + 2 * kh;
    v2f aq2[4];
#pragma unroll
    for (int k8 = 0; k8 < 4; ++k8) aq2[k8] = *(const v2f*)&xh[ar + k8 * 4];
    for (int c = 0; c < 13; ++c) {
      v8f s8 = {};
      int br = (c * 16 + lr) * 16 + 2 * kh;
#pragma unroll
      for (int k8 = 0; k8 < 4; ++k8)
        s8 = wmma4(s8, aq2[k8], *(const v2f*)&kk[br + k8 * 4]);
#pragma unroll
      for (int r = 0; r < 8; ++r)
        scw[(r + kh * 8) * 208 + c * 16 + lr] = s8[r] * 0.25f;
    }
    __builtin_amdgcn_wave_barrier();
    if (lane < 16) {  // one lane per score row
      float* rp = scw + lane * 208;
      float mx = -3.0e38f;
      for (int ci = 0; ci < S; ++ci) mx = fmaxf(mx, rp[ci]);
      float sum = 0.0f;
      for (int ci = 0; ci < S; ++ci) { float e = expf(rp[ci] - mx); rp[ci] = e; sum += e; }
      float inv = 1.0f / sum;
      for (int ci = 0; ci < S; ++ci) rp[ci] *= inv;
      for (int ci = S; ci < 208; ++ci) rp[ci] = 0.0f;
    }
    __builtin_amdgcn_wave_barrier();
    v8f o8 = {};
    const v2f* ap = (const v2f*)&scw[lr * 208 + 2 * kh];
    const float* vp = &vv[(2 * kh) * 16 + lr];
    for (int k0 = 0; k0 < 208; k0 += 4) {
      v2f b; b.x = vp[0]; b.y = vp[16];
      o8 = wmma4(o8, ap[0], b);
      ap += 2; vp += 64;
    }
#pragma unroll
    for (int r = 0; r < 8; ++r) {
      int s = t * 16 + r + kh * 8;
      if (s < S) {
        size_t oi = base + (size_t)s * 64 + lr;
        Out[oi] = o8[r] + Xres[oi];
      }
    }
  }
}

// ---------------- generic WMMA GEMM, 4 M-tiles per wave --------------------
// Requires K%4==0, N%64==0. Ragged M handled by row clamping + masked store.
__device__ inline void gemm_epi(v8f acc, int row0, int bcol, int kh, int M,
                                float bv, const float* __restrict__ Res,
                                float* __restrict__ C, int ldc, int act,
                                int depatch) {
#pragma unroll
  for (int r = 0; r < 8; ++r) {
    int crow = row0 + r + kh * 8;
    if (crow < M) {
      float v = acc[r] + bv;
      if (act == 1) v = gelu_f(v);
      if (Res != nullptr) v += Res[(size_t)crow * ldc + bcol];
      if (!depatch) {
        C[(size_t)crow * ldc + bcol] = v;
      } else {
        int nimg = crow / 196, p = crow % 196;
        int prow = (p / 14) * 32 + (bcol >> 5);
        int pcol = (p % 14) * 32 + (bcol & 31);
        C[(size_t)nimg * 200704 + (size_t)prow * 448 + pcol] = v;
      }
    }
  }
}

__global__ void gemm_wmma(const float* __restrict__ A, const float* __restrict__ B,
                          const float* __restrict__ bias, const float* __restrict__ Res,
                          float* __restrict__ C, int M, int N, int K,
                          int lda, int ldb, int ldc, int act, int depatch) {
  (void)N;
  int wave = threadIdx.x >> 5, lane = threadIdx.x & 31;
  int lr = lane & 15, kh = lane >> 4;
  int row0 = blockIdx.x * 64;
  int bcol = (blockIdx.y * 4 + wave) * 16 + lr;
  int r0 = row0 + lr;       if (r0 >= M) r0 = M - 1;   // clamp, store masked
  int r1 = row0 + 16 + lr;  if (r1 >= M) r1 = M - 1;
  int r2 = row0 + 32 + lr;  if (r2 >= M) r2 = M - 1;
  int r3 = row0 + 48 + lr;  if (r3 >= M) r3 = M - 1;
  const v2f* Ap0 = (const v2f*)(A + (size_t)r0 * lda + 2 * kh);
  const v2f* Ap1 = (const v2f*)(A + (size_t)r1 * lda + 2 * kh);
  const v2f* Ap2 = (const v2f*)(A + (size_t)r2 * lda + 2 * kh);
  const v2f* Ap3 = (const v2f*)(A + (size_t)r3 * lda + 2 * kh);
  const float* Bp = B + (size_t)(2 * kh) * ldb + bcol;
  v8f acc0 = {}, acc1 = {}, acc2 = {}, acc3 = {};
  for (int k0 = 0; k0 < K; k0 += 4) {
    v2f b; b.x = Bp[0]; b.y = Bp[ldb];
    acc0 = wmma4(acc0, Ap0[0], b);
    acc1 = wmma4(acc1, Ap1[0], b);
    acc2 = wmma4(acc2, Ap2[0], b);
    acc3 = wmma4(acc3, Ap3[0], b);
    Ap0 += 2; Ap1 += 2; Ap2 += 2; Ap3 += 2;
    Bp += (size_t)4 * ldb;
  }
  float bv = (bias != nullptr) ? bias[bcol] : 0.0f;
  gemm_epi(acc0, row0,      bcol, kh, M, bv, Res, C, ldc, act, depatch);
  gemm_epi(acc1, row0 + 16, bcol, kh, M, bv, Res, C, ldc, act, depatch);
  gemm_epi(acc2, row0 + 32, bcol, kh, M, bv, Res, C, ldc, act, depatch);
  gemm_epi(acc3, row0 + 48, bcol, kh, M, bv, Res, C, ldc, act, depatch);
}

// ---------------- bottleneck: lat = gelu(x[:,1:,:] @ W(64x1) + b) ----------
// wave-per-row dot product, class-token stripping fused into addressing.
__global__ void mlp1_k(const float* __restrict__ X, const float* __restrict__ W,
                       const float* __restrict__ b, float* __restrict__ lat) {
  int wave = threadIdx.x >> 5, lane = threadIdx.x & 31;
  int m = blockIdx.x * 8 + wave;
  if (m >= 25088) return;
  int n = m / 196, s = m % 196;
  const float* xp = X + ((size_t)n * 197 + 1 + s) * 64;
  float acc = xp[lane] * W[lane] + xp[lane + 32] * W[lane + 32];
  for (int sh = 16; sh >= 1; sh >>= 1) acc += __shfl_xor(acc, sh, 32);
  if (lane == 0) lat[m] = gelu_f(acc + b[0]);
}

// ---------------- bottleneck: x = gelu(lat @ W(1x64) + b) (outer product) --
__global__ void mlp4_k(const float* __restrict__ lat, const float* __restrict__ W,
                       const float* __restrict__ b, float* __restrict__ y) {
  int idx = blockIdx.x * blockDim.x + threadIdx.x;
  if (idx >= 25088 * 64) return;
  int m = idx >> 6, c = idx & 63;
  y[idx] = gelu_f(lat[m] * W[c] + b[c]);
}

// ---------------- add decoder positional embedding (rows 0..195) -----------
__global__ void addpos_k(float* __restrict__ y, const float* __restrict__ pos) {
  int idx = blockIdx.x * blockDim.x + threadIdx.x;
  if (idx >= 25088 * 64) return;
  int c = idx & 63, s = (idx >> 6) % 196;
  y[idx] += pos[s * 64 + c];
}

// ---------------------------------------------------------------------------
extern "C" void kernel_launch(void* const* d_in, const int* in_sizes, int n_in,
                              void* d_out, int out_size, void* d_ws, size_t ws_size,
                              hipStream_t stream) {
  (void)in_sizes; (void)n_in; (void)out_size; (void)ws_size;
  auto F = [&](int i) { return (const float*)d_in[i]; };
  const float* images = F(0);
  const float* class_token = F(1);
  const float* lm_W = F(2);  const float* lm_b = F(3);
  const float* lm2_W = F(4); const float* lm2_b = F(5);
  const float* mlp1_W = F(6); const float* mlp1_b = F(7);
  const float* mlp4_W = F(8); const float* mlp4_b = F(9);

  float* ws = (float*)d_ws;
  float* pos = ws;                        // 197*64
  float* bx  = pos + 197 * 64;            // 25216*64 token buffer (x)
  float* bh  = bx + 25216 * 64;           // residual buffer (h)
  float* bt  = bh + 25216 * 64;           // LN / temp buffer
  float* bm  = bt + 25216 * 64;           // 25216*256 MLP hidden

  posemb_k<<<dim3((197 * 64 + 255) / 256), 256, 0, stream>>>(pos);
  cls_k<<<dim3((128 * 64 + 255) / 256), 256, 0, stream>>>(bx, class_token, pos);
  patch_embed_k<<<dim3(13, 128), 128, 0, stream>>>(images, lm_W, lm_b, pos, bx);

  auto run_blocks = [&](float* x, int S, int M, int pb) {
    const float* ln1g = F(pb + 0); const float* ln1b = F(pb + 1);
    const float* ln2g = F(pb + 2); const float* ln2b = F(pb + 3);
    const float* Wq = F(pb + 4); const float* Wk = F(pb + 5); const float* Wv = F(pb + 6);
    const float* bq = F(pb + 7); const float* bk = F(pb + 8); const float* bv = F(pb + 9);
    const float* W1 = F(pb + 10); const float* b1 = F(pb + 11);
    const float* W2 = F(pb + 12); const float* b2 = F(pb + 13);
    int mt = (M + 63) / 64;
    for (int i = 0; i < 2; ++i) {
      ln_k<<<dim3((M + 7) / 8), 256, 0, stream>>>(x, ln1g + i * 64, ln1b + i * 64, bt, M);
      msa_k<<<dim3(128, 4), 128, 0, stream>>>(bt, x,
          Wq + i * 1024, Wk + i * 1024, Wv + i * 1024,
          bq + i * 64, bk + i * 64, bv + i * 64, bh, S);
      ln_k<<<dim3((M + 7) / 8), 256, 0, stream>>>(bh, ln2g + i * 64, ln2b + i * 64, bt, M);
      gemm_wmma<<<dim3(mt, 4), 128, 0, stream>>>(bt, W1 + i * 64 * 256, b1 + i * 256,
          nullptr, bm, M, 256, 64, 64, 256, 256, /*act=*/1, 0);
      gemm_wmma<<<dim3(mt, 1), 128, 0, stream>>>(bm, W2 + i * 256 * 64, b2 + i * 64,
          bh, x, M, 64, 256, 256, 64, 64, /*act=*/0, 0);
    }
  };

  // encoder (S=197, M=25216, 394 row-blocks of 64)
  run_blocks(bx, 197, 25216, 10);

  // bottleneck
  float* lat = (float*)d_out;          // first 25088 floats of output
  mlp1_k<<<dim3(25088 / 8), 256, 0, stream>>>(bx, mlp1_W, mlp1_b, lat);
  mlp4_k<<<dim3(25088 * 64 / 256), 256, 0, stream>>>(lat, mlp4_W, mlp4_b, bx);

  // decoder (S=196, M=25088, 392 row-blocks of 64)
  run_blocks(bx, 196, 25088, 24);

  // + pos emb, un-embed GEMM with fused depatchify store
  addpos_k<<<dim3(25088 * 64 / 256), 256, 0, stream>>>(bx, pos);
  float* reimg = lat + 25088;
  gemm_wmma<<<dim3(392, 16), 128, 0, stream>>>(bx, lm2_W, lm2_b, nullptr,
      reimg, 25088, 1024, 64, 64, 1024, 0, /*act=*/0, /*depatch=*/1);
}